// VanillaAttention_49143015801155
// MI455X (gfx1250) — compile-verified
//
#include <hip/hip_runtime.h>

#define B_  4
#define TC  16
#define E_  4096
#define H_  32
#define HD_ 128
#define TP  8192
#define TT  (TP + TC)   // 8208

typedef float v2f __attribute__((ext_vector_type(2)));
typedef float v8f __attribute__((ext_vector_type(8)));

__device__ __forceinline__ v8f wmma4(v2f a, v2f b, v8f c) {
  // D = A(16x4) * B(4x16) + C(16x16), fp32 WMMA
  return __builtin_amdgcn_wmma_f32_16x16x4_f32(false, a, false, b, (short)0, c,
                                               false, false);
}

__device__ __forceinline__ v2f ld2(const float* p) {
  float2 t = *reinterpret_cast<const float2*>(p);
  v2f r; r[0] = t.x; r[1] = t.y; return r;
}

// One async 16B-per-lane copy: 512B per instruction, tracked on ASYNCcnt.
__device__ __forceinline__ void async_copy16(unsigned ldsaddr,
                                             unsigned long long gaddr) {
  asm volatile("global_load_async_to_lds_b128 %0, %1, off"
               :
               : "v"(ldsaddr), "v"(gaddr)
               : "memory");
}

__device__ __forceinline__ void wait_async0() {
  asm volatile("s_wait_asynccnt 0" ::: "memory");
}

// ---------------------------------------------------------------------------
// Kernel 1: copy K_prev/V_prev into the concat cache layout (float4 streams)
// ---------------------------------------------------------------------------
__global__ __launch_bounds__(256) void copy_prev_kernel(
    const float4* __restrict__ kprev, const float4* __restrict__ vprev,
    float4* __restrict__ kc, float4* __restrict__ vc) {
  const size_t N4   = (size_t)B_ * H_ * TP * HD_ / 4;  // per tensor
  const size_t srcC = (size_t)TP * HD_ / 4;            // per (b,h) source
  const size_t dstC = (size_t)TT * HD_ / 4;            // per (b,h) dest
  for (size_t i = (size_t)blockIdx.x * blockDim.x + threadIdx.x; i < N4;
       i += (size_t)gridDim.x * blockDim.x) {
    size_t bh = i / srcC;
    size_t r  = i - bh * srcC;
    kc[bh * dstC + r] = kprev[i];
    vc[bh * dstC + r] = vprev[i];
  }
}

// ---------------------------------------------------------------------------
// Kernel 2: q/k/v = x @ W^T + b via f32 WMMA. One wave = one projection x one
// 16-wide N stripe x all 4 M-tiles -> every weight read exactly once.
// ---------------------------------------------------------------------------
__global__ __launch_bounds__(256) void proj_qkv_kernel(
    const float* __restrict__ x,
    const float* __restrict__ Wq, const float* __restrict__ bq,
    const float* __restrict__ Wk, const float* __restrict__ bk,
    const float* __restrict__ Wv, const float* __restrict__ bv,
    float* __restrict__ ws_q, float* __restrict__ kcache,
    float* __restrict__ vcache) {
  const int lane = threadIdx.x & 31;
  const int wv   = threadIdx.x >> 5;
  const int wg   = blockIdx.x * 8 + wv;   // 0..767
  const int proj = wg >> 8;               // 0=q 1=k 2=v
  const int nt   = wg & 255;
  const int n0   = nt * 16;
  const int lr   = lane & 15;
  const int hk   = lane >> 4;             // 0: K=k0,k0+1  1: K=k0+2,k0+3

  const float* W    = (proj == 0) ? Wq : (proj == 1) ? Wk : Wv;
  const float* bias = (proj == 0) ? bq : (proj == 1) ? bk : bv;

  const float* wrow = W + (size_t)(n0 + lr) * E_ + hk * 2;  // B = W^T column
  const float* xrow = x + (size_t)lr * E_ + hk * 2;         // A rows

  v8f acc0 = {}, acc1 = {}, acc2 = {}, acc3 = {};
  for (int k = 0; k < E_; k += 4) {
    v2f bfrag = ld2(wrow + k);
    v2f a0 = ld2(xrow + k);
    v2f a1 = ld2(xrow + 16 * E_ + k);
    v2f a2 = ld2(xrow + 32 * E_ + k);
    v2f a3 = ld2(xrow + 48 * E_ + k);
    acc0 = wmma4(a0, bfrag, acc0);
    acc1 = wmma4(a1, bfrag, acc1);
    acc2 = wmma4(a2, bfrag, acc2);
    acc3 = wmma4(a3, bfrag, acc3);
  }

  const float bval = bias[n0 + lr];
  const int col = n0 + lr;
  const int h = col >> 7;
  const int d = col & 127;
  v8f accs[4] = {acc0, acc1, acc2, acc3};
#pragma unroll
  for (int mt = 0; mt < 4; ++mt) {
#pragma unroll
    for (int j = 0; j < 8; ++j) {
      float val = accs[mt][j] + bval;
      int t  = j + hk * 8;   // time index within the 16-row tile
      int bb = mt;           // batch index (M = bb*16 + t)
      if (proj == 0) {
        ws_q[(((size_t)bb * H_ + h) * TC + t) * HD_ + d] = val;
      } else if (proj == 1) {
        kcache[(((size_t)bb * H_ + h) * TT + TP + t) * HD_ + d] = val;
      } else {
        vcache[(((size_t)bb * H_ + h) * TT + TP + t) * HD_ + d] = val;
      }
    }
  }
}

// ---------------------------------------------------------------------------
// Kernel 3: flash attention, async-LDS double-buffered KV stream.
// One workgroup (8 waves) per (b,h). Each chunk (16 timesteps, 8KB K + 8KB V)
// is async-copied into shared double buffers (coalesced B128, ASYNCcnt).
// All 8 waves compute the same S tile + online softmax (cheap, keeps stats
// globally correct without any merge); wave w owns d-tile w of O += P @ V.
// ---------------------------------------------------------------------------
__global__ __launch_bounds__(256) void attn_kernel(
    const float* __restrict__ ws_q,
    const float* __restrict__ kprev, const float* __restrict__ vprev,
    const float* __restrict__ kcache, const float* __restrict__ vcache,
    float* __restrict__ ws_ctx) {
  __shared__ float sK[2][16 * HD_];   // 2 x 8KB
  __shared__ float sV[2][16 * HD_];   // 2 x 8KB
  __shared__ float sP[8][256];        // per-wave P tile (16x16)

  const int lane = threadIdx.x & 31;
  const int w    = threadIdx.x >> 5;  // 0..7, owns d-tile w
  const int bh   = blockIdx.x;        // 0..127
  const int b    = bh >> 5;
  const int h    = bh & 31;
  const int lr   = lane & 15;
  const int hk   = lane >> 4;

  const v8f vzero = {};

  // q in A-fragment layout, pre-scaled by 1/sqrt(HD)
  const float scale = 0.088388347648318447f;
  const float* qb = ws_q + ((size_t)bh * TC + lr) * HD_ + hk * 2;
  v2f aq[32];
#pragma unroll
  for (int i = 0; i < 32; ++i) {
    v2f t = ld2(qb + 4 * i);
    aq[i][0] = t[0] * scale;
    aq[i][1] = t[1] * scale;
  }

  const float* kp = kprev + (size_t)bh * TP * HD_;
  const float* vp = vprev + (size_t)bh * TP * HD_;
  const float* kt = kcache + ((size_t)bh * TT + TP) * HD_;  // new 16 rows
  const float* vt = vcache + ((size_t)bh * TT + TP) * HD_;

  const unsigned ldsK0 = (unsigned)(size_t)(void*)&sK[0][0];
  const unsigned ldsV0 = (unsigned)(size_t)(void*)&sV[0][0];

  float m[8], l[8];
  v8f O = vzero;
#pragma unroll
  for (int j = 0; j < 8; ++j) { m[j] = -3.0e38f; l[j] = 0.0f; }

  const int nchunks = TT / 16;  // 513

  // Prologue: issue chunk 0 into buffer 0. This wave copies segments
  // {2w, 2w+1} of both K and V (16 segments x 512B = 8KB each).
  {
#pragma unroll
    for (int j = 0; j < 2; ++j) {
      const int off = (w * 2 + j) * 512 + lane * 16;
      async_copy16(ldsK0 + off, (unsigned long long)((const char*)kp + off));
      async_copy16(ldsV0 + off, (unsigned long long)((const char*)vp + off));
    }
  }

  for (int c = 0; c < nchunks; ++c) {
    const int buf = c & 1;
    wait_async0();       // my async copies for chunk c have landed
    __syncthreads();     // everyone's copies landed; prev buffer fully read

    // Issue chunk c+1 into the other buffer while we compute chunk c.
    if (c + 1 < nchunks) {
      const unsigned bo = ((unsigned)((c + 1) & 1)) * (16 * HD_ * 4);
      const char* gk = (c + 1 < TP / 16)
                           ? (const char*)(kp + (size_t)(c + 1) * 16 * HD_)
                           : (const char*)kt;
      const char* gv = (c + 1 < TP / 16)
                           ? (const char*)(vp + (size_t)(c + 1) * 16 * HD_)
                           : (const char*)vt;
#pragma unroll
      for (int j = 0; j < 2; ++j) {
        const int off = (w * 2 + j) * 512 + lane * 16;
        async_copy16(ldsK0 + bo + off, (unsigned long long)(gk + off));
        async_copy16(ldsV0 + bo + off, (unsigned long long)(gv + off));
      }
    }

    const float* kbuf = &sK[buf][0];
    const float* vbuf = &sV[buf][0];

    // S = q @ K_chunk^T from LDS (B fragment = contiguous float2 of a K row)
    v8f s = vzero;
    const float* kr = kbuf + lr * HD_ + hk * 2;
#pragma unroll
    for (int i = 0; i < 32; ++i) {
      v2f bf = ld2(kr + 4 * i);
      s = wmma4(aq[i], bf, s);
    }

    // online softmax per row fragment (identical in all 8 waves)
#pragma unroll
    for (int j = 0; j < 8; ++j) {
      float sj = s[j];
      float rmax = sj;
      rmax = fmaxf(rmax, __shfl_xor(rmax, 1, 32));
      rmax = fmaxf(rmax, __shfl_xor(rmax, 2, 32));
      rmax = fmaxf(rmax, __shfl_xor(rmax, 4, 32));
      rmax = fmaxf(rmax, __shfl_xor(rmax, 8, 32));
      float mn    = fmaxf(m[j], rmax);
      float alpha = __expf(m[j] - mn);
      float p     = __expf(sj - mn);
      float ps = p;
      ps += __shfl_xor(ps, 1, 32);
      ps += __shfl_xor(ps, 2, 32);
      ps += __shfl_xor(ps, 4, 32);
      ps += __shfl_xor(ps, 8, 32);
      l[j] = l[j] * alpha + ps;
      m[j] = mn;
      O[j] *= alpha;
      sP[w][(j + hk * 8) * 16 + lr] = p;  // row-major P tile (wave-private)
    }
    __builtin_amdgcn_wave_barrier();

    // O += P @ V_chunk for this wave's d-tile (columns w*16 .. w*16+15)
    const int colv = w * 16 + lr;
#pragma unroll
    for (int ks = 0; ks < 16; ks += 4) {
      v2f pa = ld2(&sP[w][lr * 16 + ks + hk * 2]);
      const float* vr = vbuf + (size_t)(ks + hk * 2) * HD_ + colv;
      v2f bf;
      bf[0] = vr[0];
      bf[1] = vr[HD_];
      O = wmma4(pa, bf, O);
    }
    // next iteration's __syncthreads guards the buffer swap
  }

  // Every wave saw every chunk -> m/l are final; write own d-tile.
#pragma unroll
  for (int j = 0; j < 8; ++j) {
    const int row = j + hk * 8;
    ws_ctx[((size_t)b * TC + row) * E_ + (size_t)h * HD_ + w * 16 + lr] =
        O[j] / l[j];
  }
}

// ---------------------------------------------------------------------------
// Kernel 4: out = context @ Wo^T + bo
// ---------------------------------------------------------------------------
__global__ __launch_bounds__(256) void out_proj_kernel(
    const float* __restrict__ ctx, const float* __restrict__ Wo,
    const float* __restrict__ bo, float* __restrict__ out) {
  const int lane = threadIdx.x & 31;
  const int wv   = threadIdx.x >> 5;
  const int nt   = blockIdx.x * 8 + wv;  // 0..255
  const int n0   = nt * 16;
  const int lr   = lane & 15;
  const int hk   = lane >> 4;

  const float* wrow = Wo + (size_t)(n0 + lr) * E_ + hk * 2;
  const float* xrow = ctx + (size_t)lr * E_ + hk * 2;

  v8f acc0 = {}, acc1 = {}, acc2 = {}, acc3 = {};
  for (int k = 0; k < E_; k += 4) {
    v2f bf = ld2(wrow + k);
    acc0 = wmma4(ld2(xrow + k),           bf, acc0);
    acc1 = wmma4(ld2(xrow + 16 * E_ + k), bf, acc1);
    acc2 = wmma4(ld2(xrow + 32 * E_ + k), bf, acc2);
    acc3 = wmma4(ld2(xrow + 48 * E_ + k), bf, acc3);
  }
  const float bval = bo[n0 + lr];
  v8f accs[4] = {acc0, acc1, acc2, acc3};
#pragma unroll
  for (int mt = 0; mt < 4; ++mt)
#pragma unroll
    for (int j = 0; j < 8; ++j) {
      int mrow = mt * 16 + j + hk * 8;
      out[(size_t)mrow * E_ + n0 + lr] = accs[mt][j] + bval;
    }
}

// ---------------------------------------------------------------------------
extern "C" void kernel_launch(void* const* d_in, const int* in_sizes, int n_in,
                              void* d_out, int out_size, void* d_ws,
                              size_t ws_size, hipStream_t stream) {
  (void)in_sizes; (void)n_in; (void)out_size; (void)ws_size;

  const float* x  = (const float*)d_in[0];
  const float* Kp = (const float*)d_in[1];
  const float* Vp = (const float*)d_in[2];
  const float* Wq = (const float*)d_in[3];
  const float* bq = (const float*)d_in[4];
  const float* Wk = (const float*)d_in[5];
  const float* bk = (const float*)d_in[6];
  const float* Wv = (const float*)d_in[7];
  const float* bv = (const float*)d_in[8];
  const float* Wo = (const float*)d_in[9];
  const float* bo = (const float*)d_in[10];

  float* out    = (float*)d_out;                         // [64, 4096]
  float* kcache = out + (size_t)B_ * TC * E_;            // [B,H,TT,HD]
  float* vcache = kcache + (size_t)B_ * H_ * TT * HD_;   // [B,H,TT,HD]

  float* ws_q   = (float*)d_ws;                          // [B,H,TC,HD] 1MB
  float* ws_ctx = ws_q + (size_t)B_ * H_ * TC * HD_;     // [B,TC,E]    1MB

  copy_prev_kernel<<<dim3(8192), dim3(256), 0, stream>>>(
      (const float4*)Kp, (const float4*)Vp, (float4*)kcache, (float4*)vcache);
  proj_qkv_kernel<<<dim3(96), dim3(256), 0, stream>>>(
      x, Wq, bq, Wk, bk, Wv, bv, ws_q, kcache, vcache);
  attn_kernel<<<dim3(128), dim3(256), 0, stream>>>(
      ws_q, Kp, Vp, kcache, vcache, ws_ctx);
  out_proj_kernel<<<dim3(32), dim3(256), 0, stream>>>(ws_ctx, Wo, bo, out);
}